// SDF_75806172774865
// MI455X (gfx1250) — compile-verified
//
#include <hip/hip_runtime.h>

#define TPB   256   // threads per block = points per block
#define KC    4     // k's staged per chunk (K must be a multiple of 4; K=60 here)
#define BATCH 7     // async-copy instructions issued per wave per chunk (6 v/vn + 1 sr)

// ---------------------------------------------------------------------------
// CDNA5 async global->LDS copy (ASYNCcnt path). 16 bytes per lane per call.
// Probe-verified signature: (v4i as1* src, v4i as3* dst, imm offset, imm cpol)
// ---------------------------------------------------------------------------
typedef __attribute__((ext_vector_type(4))) int v4i;
typedef __attribute__((address_space(1))) v4i g_v4i;   // global
typedef __attribute__((address_space(3))) v4i l_v4i;   // LDS

__device__ __forceinline__ void async_cp16(const float* src, float* dst) {
#if __has_builtin(__builtin_amdgcn_global_load_async_to_lds_b128)
    __builtin_amdgcn_global_load_async_to_lds_b128(
        (g_v4i*)src, (l_v4i*)dst, /*offset=*/0, /*cpol=*/0);
#else
    *(float4*)dst = *(const float4*)src;   // synchronous fallback
#endif
}

template <int CNT>
__device__ __forceinline__ void async_wait_le() {
#if __has_builtin(__builtin_amdgcn_s_wait_asynccnt)
    __builtin_amdgcn_s_wait_asynccnt(CNT);
#elif __has_builtin(__builtin_amdgcn_global_load_async_to_lds_b128)
    asm volatile("s_wait_asynccnt %0" ::"i"(CNT) : "memory");
#endif
}

// ---------------------------------------------------------------------------
// Kernel 1: per-block partial sums of sdf^2. grid=(nblocks,N), block=TPB.
// Double-buffered async global->LDS staging of (v, vn, sr) K-chunks.
// ---------------------------------------------------------------------------
__global__ void __launch_bounds__(TPB)
sdf_partial_kernel(const float* __restrict__ points,
                   const float* __restrict__ v,
                   const float* __restrict__ vn,
                   const float* __restrict__ sr,
                   float* __restrict__ partial,
                   int P, int K) {
    __shared__ __align__(16) float s_v [2][TPB * KC * 3];  // 12 floats / point
    __shared__ __align__(16) float s_vn[2][TPB * KC * 3];
    __shared__ __align__(16) float s_sr[2][TPB * KC];      // 4 floats / point
    __shared__ float s_red[TPB];

    const int n  = blockIdx.y;
    const int t  = threadIdx.x;
    const int p0 = blockIdx.x * TPB;
    const int p  = p0 + t;
    const bool active = (p < P);
    const int  pc = active ? p : (P - 1);              // clamped point for loads

    const long long npBase = (long long)n * P;

    // this thread's query point
    const long long pb = (npBase + pc) * 3;
    const float px = points[pb + 0];
    const float py = points[pb + 1];
    const float pz = points[pb + 2];

    const int nchunks    = K / KC;
    const int vf4_per_pt = (K * 3) >> 2;   // float4s per point in v/vn (45)
    const int sf4_per_pt = K >> 2;         // float4s per point in sr  (15)

    // ---- hoisted staging assignments: 3 (pt,j) slots per thread -----------
    // flat index f = t + i*TPB -> pt = f/3, j = f%3. Chunk c adds c*3 float4s.
    const float* gv [3];
    const float* gvn[3];
    int          loff[3];
#pragma unroll
    for (int i = 0; i < 3; ++i) {
        const int f  = t + i * TPB;
        const int pt = f / 3;
        const int j  = f - pt * 3;
        int pg = p0 + pt; if (pg > P - 1) pg = P - 1;
        const long long g = (npBase + pg) * (long long)vf4_per_pt + j;  // float4 idx
        gv [i]  = v  + g * 4;
        gvn[i]  = vn + g * 4;
        loff[i] = pt * (KC * 3) + j * 4;
    }
    const float* gsr = sr + ((npBase + pc) * (long long)sf4_per_pt) * 4;

    // per-chunk strides (floats): v/vn advance 3 float4s, sr advances 1
    #define STAGE(buf, c)                                                     \
        do {                                                                  \
            _Pragma("unroll")                                                 \
            for (int i = 0; i < 3; ++i) {                                     \
                async_cp16(gv [i] + (c) * 12, &s_v [(buf)][loff[i]]);         \
                async_cp16(gvn[i] + (c) * 12, &s_vn[(buf)][loff[i]]);         \
            }                                                                 \
            async_cp16(gsr + (c) * 4, &s_sr[(buf)][t * KC]);                  \
        } while (0)

    float num = 0.0f;
    float den = 0.0f;

    STAGE(0, 0);                       // prologue: chunk 0 into buffer 0

    for (int c = 0; c < nchunks; ++c) {
        const int cur = c & 1;
        if (c + 1 < nchunks) {
            STAGE(cur ^ 1, c + 1);     // prefetch next chunk into other buffer
            async_wait_le<BATCH>();    // retire chunk c's batch; keep c+1 inflight
        } else {
            async_wait_le<0>();        // last chunk: drain
        }
        __syncthreads();               // chunk c visible block-wide

        // ---- consume KC k-values for this thread's point (b128 LDS reads) --
        const float4 a0 = ((const float4*)&s_v [cur][t * 12])[0];
        const float4 a1 = ((const float4*)&s_v [cur][t * 12])[1];
        const float4 a2 = ((const float4*)&s_v [cur][t * 12])[2];
        const float4 b0 = ((const float4*)&s_vn[cur][t * 12])[0];
        const float4 b1 = ((const float4*)&s_vn[cur][t * 12])[1];
        const float4 b2 = ((const float4*)&s_vn[cur][t * 12])[2];
        const float4 rr = ((const float4*)&s_sr[cur][t * 4])[0];

        const float vx[KC] = {a0.x, a0.w, a1.z, a2.y};
        const float vy[KC] = {a0.y, a1.x, a1.w, a2.z};
        const float vz[KC] = {a0.z, a1.y, a2.x, a2.w};
        const float nx[KC] = {b0.x, b0.w, b1.z, b2.y};
        const float ny[KC] = {b0.y, b1.x, b1.w, b2.z};
        const float nz[KC] = {b0.z, b1.y, b2.x, b2.w};
        const float rv[KC] = {rr.x, rr.y, rr.z, rr.w};

#pragma unroll
        for (int kk = 0; kk < KC; ++kk) {
            const float dx = px - vx[kk];
            const float dy = py - vy[kk];
            const float dz = pz - vz[kk];
            const float dist = fmaf(dx, dx, fmaf(dy, dy, dz * dz));

            const float w   = 1.0f - dist / rv[kk];
            const float w2  = w * w;
            const float phi = (dist < rv[kk]) ? (w2 * w2) : 1e-18f;

            const float dotn = fmaf(nx[kk], dx, fmaf(ny[kk], dy, nz[kk] * dz));
            num = fmaf(phi, dotn, num);
            den += phi;
        }
        __syncthreads();               // buffer cur reusable for chunk c+2
    }
    #undef STAGE

    const float sdf = num / den;
    s_red[t] = active ? (sdf * sdf) : 0.0f;
    __syncthreads();
#pragma unroll
    for (int s = TPB / 2; s > 0; s >>= 1) {
        if (t < s) s_red[t] += s_red[t + s];
        __syncthreads();
    }
    if (t == 0) partial[(long long)n * gridDim.x + blockIdx.x] = s_red[0];
}

// ---------------------------------------------------------------------------
// Kernel 2: deterministic reduction of the per-block partials. grid = N.
// ---------------------------------------------------------------------------
__global__ void __launch_bounds__(TPB)
reduce_partials_kernel(const float* __restrict__ partial,
                       float* __restrict__ out, int nblocks) {
    __shared__ float s[TPB];
    const int n = blockIdx.x;
    const int t = threadIdx.x;
    float acc = 0.0f;
    for (int i = t; i < nblocks; i += TPB)
        acc += partial[(long long)n * nblocks + i];
    s[t] = acc;
    __syncthreads();
#pragma unroll
    for (int sh = TPB / 2; sh > 0; sh >>= 1) {
        if (t < sh) s[t] += s[t + sh];
        __syncthreads();
    }
    if (t == 0) out[n] = s[0];
}

// ---------------------------------------------------------------------------
extern "C" void kernel_launch(void* const* d_in, const int* in_sizes, int n_in,
                              void* d_out, int out_size, void* d_ws, size_t ws_size,
                              hipStream_t stream) {
    const float* points = (const float*)d_in[0];   // (N,P,3)
    const float* v      = (const float*)d_in[1];   // (N,P,K,3)
    const float* vn     = (const float*)d_in[2];   // (N,P,K,3)
    const float* sr     = (const float*)d_in[3];   // (N,P,K)
    float* out = (float*)d_out;                    // (N,)

    const int N = out_size;                                   // 2
    const int P = (int)((long long)in_sizes[0] / (3LL * N));  // 100000
    const int K = (int)((long long)in_sizes[3] / ((long long)N * P)); // 60

    const int nblocks = (P + TPB - 1) / TPB;
    float* partial = (float*)d_ws;                 // N * nblocks floats

    dim3 grid(nblocks, N);
    sdf_partial_kernel<<<grid, TPB, 0, stream>>>(points, v, vn, sr, partial, P, K);
    reduce_partials_kernel<<<N, TPB, 0, stream>>>(partial, out, nblocks);
}